// STS3M_58308476011159
// MI455X (gfx1250) — compile-verified
//
#include <hip/hip_runtime.h>
#include <hip/hip_bf16.h>

// ---------------------------------------------------------------------------
// STS3M graph-Mamba forward for MI455X (gfx1250, wave32).
// fp32 WMMA GEMMs with TDM (tensor_load_to_lds) weight staging, fused
// elementwise epilogues, collapsed delta@pad, register-resident SSM scan.
// ---------------------------------------------------------------------------

typedef __attribute__((ext_vector_type(2))) float v2f;
typedef __attribute__((ext_vector_type(8))) float v8f;
typedef __attribute__((ext_vector_type(4))) unsigned int u32x4;
typedef __attribute__((ext_vector_type(8))) int i32x8;
typedef __attribute__((ext_vector_type(4))) int i32x4;

#define R_TOT   16384      // B*N*T rows
#define DIN     512        // d_inner
#define DST     16         // d_state
#define DTR     32         // dt_rank
#define FIN     64
#define FOUT    64
#define TLEN    128
#define NSEQ    128        // B*N

__device__ __forceinline__ float siluf(float x) {
    return x / (1.f + __expf(-x));
}

// ---------------------------------------------------------------------------
// TDM: DMA a K x 16 fp32 column-slice of a row-major (tensor_h x tensor_w)
// matrix into LDS (contiguous 16-float rows). D# built per CDNA5 ISA 8.3/8.4.
// Issued once per wave (TDM is a wave-level op); caller gates to wave 0.
// ---------------------------------------------------------------------------
__device__ __forceinline__ void tdm_load_slice(const float* gptr, unsigned lds_off,
                                               int tensor_w, int tensor_h,
                                               int tile_w, int tile_h) {
    const unsigned long long ga = (unsigned long long)(uintptr_t)gptr;
    u32x4 g0;
    g0.x = 1u;                                            // count=1, flags 0
    g0.y = lds_off;                                       // lds_addr (bytes)
    g0.z = (unsigned)(ga & 0xFFFFFFFFu);                  // global_addr[31:0]
    g0.w = (unsigned)((ga >> 32) & 0x1FFFFFFu) | (2u << 30);  // [56:32] | type=2
    i32x8 g1;
    g1[0] = (2 << 16);                                    // data_size = 4B
    g1[1] = (tensor_w & 0xFFFF) << 16;                    // tensor_dim0[15:0]
    g1[2] = ((tensor_w >> 16) & 0xFFFF) | ((tensor_h & 0xFFFF) << 16);
    g1[3] = ((tensor_h >> 16) & 0xFFFF) | ((tile_w & 0xFFFF) << 16); // tile_dim0
    g1[4] = (tile_h & 0xFFFF);                            // tile_dim1; tile_dim2=0
    g1[5] = tensor_w;                                     // tensor_dim0_stride[31:0]
    g1[6] = 0;                                            // stride[47:32] | dim1_stride lo
    g1[7] = 0;
    i32x4 z4 = {0, 0, 0, 0};
    i32x8 z8 = {0, 0, 0, 0, 0, 0, 0, 0};
    // amdgpu-toolchain (clang-23) 6-arg form
    __builtin_amdgcn_tensor_load_to_lds(g0, g1, z4, z4, z8, 0);
}

// One wave: 16x16 C tile of A(MxK row-major, global) @ B(Kx16, staged in LDS).
// fp32 WMMA operand layout per CDNA5 ISA 7.12.2.
__device__ __forceinline__ v8f wmma_tile_ldsB(const float* __restrict__ A, int lda,
                                              const float* sB, int K, int lane) {
    v8f acc = {};
    const int ln = lane & 15;
    const int kk = (lane >> 4) * 2;
    for (int k0 = 0; k0 < K; k0 += 4) {
        v2f a, b;
        a.x = A[ln * lda + k0 + kk];
        a.y = A[ln * lda + k0 + kk + 1];
        b.x = sB[(k0 + kk) * 16 + ln];
        b.y = sB[(k0 + kk + 1) * 16 + ln];
        acc = __builtin_amdgcn_wmma_f32_16x16x4_f32(false, a, false, b,
                                                    (short)0, acc, false, false);
    }
    return acc;
}

// Global-B variant (used by k_delta where B is tiny and bias fused).
__device__ __forceinline__ v8f wmma_f32_tile(const float* __restrict__ A, int lda,
                                             const float* __restrict__ B, int ldb,
                                             int K, int lane) {
    v8f acc = {};
    const int ln = lane & 15;
    const int kk = (lane >> 4) * 2;
    for (int k0 = 0; k0 < K; k0 += 4) {
        v2f a, b;
        a.x = A[ln * lda + k0 + kk];
        a.y = A[ln * lda + k0 + kk + 1];
        b.x = B[(k0 + kk) * ldb + ln];
        b.y = B[(k0 + kk + 1) * ldb + ln];
        acc = __builtin_amdgcn_wmma_f32_16x16x4_f32(false, a, false, b,
                                                    (short)0, acc, false, false);
    }
    return acc;
}

// ---------------------------------------------------------------------------
// Block-cooperative staged GEMM: all 8 waves share one N-tile (tn); TDM stages
// the Kx16 B-slice to LDS once. EPI=0: C = A@B. EPI=1: gemm_in epilogue
// (cols <512 -> xc raw; cols >=512 -> silu -> sres).
// ---------------------------------------------------------------------------
template <int K, int N, int EPI>
__global__ void k_gemm_tdm(const float* __restrict__ A, const float* __restrict__ B,
                           float* __restrict__ C, float* __restrict__ C2) {
    __shared__ float sB[K * 16];
    constexpr int TILES_N = N / 16;
    const int tn = blockIdx.x % TILES_N;
    const int tmBase = (blockIdx.x / TILES_N) * 8;
    const int tid = threadIdx.x;
    const int lane = tid & 31;
    const int wv = tid >> 5;

    if (wv == 0) {                      // wave 0 issues the DMA for the block
        const unsigned lds_off = (unsigned)(unsigned long long)(uintptr_t)&sB[0];
        tdm_load_slice(B + tn * 16, lds_off, N, K, 16, K);
        __builtin_amdgcn_s_wait_tensorcnt((short)0);
    }
    __syncthreads();

    const int tm = tmBase + wv;
    v8f acc = wmma_tile_ldsB(A + tm * 16 * K, K, sB, K, lane);

    const int n  = tn * 16 + (lane & 15);
    const int m0 = tm * 16 + (lane >> 4) * 8;
    if (EPI == 0) {
#pragma unroll
        for (int i = 0; i < 8; ++i) C[(m0 + i) * N + n] = acc[i];
    } else {
#pragma unroll
        for (int i = 0; i < 8; ++i) {
            const float v = acc[i];
            const int row = m0 + i;
            if (n < DIN) C[row * DIN + n] = v;                  // xc
            else         C2[row * DIN + (n - DIN)] = siluf(v);  // silu(res)
        }
    }
}

// ---------------- causal depthwise conv (k=4) + bias + SiLU -----------------
__global__ void k_conv_silu(const float* __restrict__ xc, const float* __restrict__ Wc,
                            const float* __restrict__ bc, float* __restrict__ u) {
    const int idx = blockIdx.x * blockDim.x + threadIdx.x;
    const int r = idx >> 9;
    const int d = idx & (DIN - 1);
    const int t = r & (TLEN - 1);
    float acc = bc[d];
#pragma unroll
    for (int k = 0; k < 4; ++k) {
        const int tt = t - 3 + k;
        if (tt >= 0) acc += xc[(r - 3 + k) * DIN + d] * Wc[d * 4 + k];
    }
    u[idx] = siluf(acc);
}

// ---------------- delta = softplus(x_dbl[:,:32]@W_dt + b_dt), collapsed dp --
__global__ void k_delta(const float* __restrict__ xdbl, const float* __restrict__ Wdt,
                        const float* __restrict__ bdt, const int* __restrict__ adj,
                        float* __restrict__ dp) {
    __shared__ float sd[16][DIN + 8];
    __shared__ float adjf[32][32];
    __shared__ float red[16][16], redh[16][16];
    __shared__ float sall[16], shead[16];
    const int tid  = threadIdx.x;
    const int lane = tid & 31;
    const int wv   = tid >> 5;
    const int row0 = blockIdx.x * 16;

    for (int i = tid; i < 1024; i += 256) adjf[i >> 5][i & 31] = (float)adj[i];

    const float* A = xdbl + row0 * 64;
    for (int nt = wv; nt < DIN / 16; nt += 8) {
        v8f acc = wmma_f32_tile(A, 64, Wdt + nt * 16, DIN, DTR, lane);
        const int n  = nt * 16 + (lane & 15);
        const float bb = bdt[n];
        const int mb = (lane >> 4) * 8;
#pragma unroll
        for (int i = 0; i < 8; ++i) {
            const float v = acc[i] + bb;
            sd[mb + i][n] = (v > 20.f) ? v : logf(1.f + __expf(v));
        }
    }
    __syncthreads();

    const int r = tid >> 4, c = tid & 15;
    float p = 0.f;
    for (int j = c; j < DIN; j += 16) p += sd[r][j];
    red[r][c]  = p;
    redh[r][c] = sd[r][c] + sd[r][c + 16];
    __syncthreads();
    if (c == 0) {
        float s = 0.f, sh = 0.f;
#pragma unroll
        for (int j = 0; j < 16; ++j) { s += red[r][j]; sh += redh[r][j]; }
        sall[r] = s; shead[r] = sh;
    }
    __syncthreads();

    for (int i = 0; i < DIN / 16; ++i) {
        const int j = c + 16 * i;
        float v;
        if (j < 32) {
            float a2 = sall[r] - shead[r];
#pragma unroll
            for (int d = 0; d < 32; ++d) a2 += sd[r][d] * adjf[d][j];
            v = a2;
        } else {
            v = sall[r];
        }
        dp[(row0 + r) * DIN + j] = v;
    }
}

// ---------------- sequential SSM scan, state in registers -------------------
__global__ void k_scan(const float* __restrict__ dp, const float* __restrict__ uu,
                       const float* __restrict__ xdbl, const float* __restrict__ sres,
                       const float* __restrict__ A_log, const float* __restrict__ Dv,
                       float* __restrict__ yact) {
    __shared__ float sB[DST], sC[DST];
    const int tid = threadIdx.x;
    const int seq = blockIdx.x;
    const int d0  = tid * 2;
    float a0[DST], a1[DST], h0[DST], h1[DST];
#pragma unroll
    for (int n = 0; n < DST; ++n) {
        a0[n] = -__expf(A_log[d0 * DST + n]);
        a1[n] = -__expf(A_log[(d0 + 1) * DST + n]);
        h0[n] = 0.f; h1[n] = 0.f;
    }
    const float D0 = Dv[d0], D1 = Dv[d0 + 1];
    const int rbase = seq * TLEN;

    for (int t = 0; t < TLEN; ++t) {
        const int r = rbase + t;
        __syncthreads();
        if (tid < 32) {
            const float v = xdbl[r * 64 + DTR + tid];
            if (tid < DST) sB[tid] = v; else sC[tid - DST] = v;
        }
        __syncthreads();
        const float dp0 = dp[r * DIN + d0], dp1 = dp[r * DIN + d0 + 1];
        const float u0 = uu[r * DIN + d0],  u1 = uu[r * DIN + d0 + 1];
        const float du0 = dp0 * u0, du1 = dp1 * u1;
        float y0 = 0.f, y1 = 0.f;
#pragma unroll
        for (int n = 0; n < DST; ++n) {
            h0[n] = __expf(dp0 * a0[n]) * h0[n] + du0 * sB[n];
            y0 += h0[n] * sC[n];
            h1[n] = __expf(dp1 * a1[n]) * h1[n] + du1 * sB[n];
            y1 += h1[n] * sC[n];
        }
        yact[r * DIN + d0]     = (y0 + u0 * D0) * sres[r * DIN + d0];
        yact[r * DIN + d0 + 1] = (y1 + u1 * D1) * sres[r * DIN + d0 + 1];
        if (t + 1 < TLEN) {
            __builtin_prefetch(&dp[(r + 1) * DIN + d0], 0, 1);
            __builtin_prefetch(&uu[(r + 1) * DIN + d0], 0, 1);
        }
    }
}

// ---------------------------------------------------------------------------
extern "C" void kernel_launch(void* const* d_in, const int* in_sizes, int n_in,
                              void* d_out, int out_size, void* d_ws, size_t ws_size,
                              hipStream_t stream) {
    const float* x      = (const float*)d_in[0];
    const int*   adj    = (const int*)  d_in[1];
    const float* W_in   = (const float*)d_in[2];
    const float* W_conv = (const float*)d_in[3];
    const float* b_conv = (const float*)d_in[4];
    const float* W_xprj = (const float*)d_in[5];
    const float* W_dt   = (const float*)d_in[6];
    const float* b_dt   = (const float*)d_in[7];
    const float* A_log  = (const float*)d_in[8];
    const float* Dv     = (const float*)d_in[9];
    const float* W_out  = (const float*)d_in[10];
    float* out = (float*)d_out;

    const size_t S = (size_t)R_TOT * DIN;          // 8.4M floats (32 MB)
    float* ws    = (float*)d_ws;
    float* xc    = ws;                             // dead after conv -> reused
    float* sres  = ws + S;
    float* u     = ws + 2 * S;
    float* xdbl  = ws + 3 * S;                     // R_TOT*64 floats
    float* dp    = ws + 3 * S + (size_t)R_TOT * 64;
    float* yact  = xc;

    // 1) xz = x @ W_in   (64 N-tiles x 128 M-groups = 8192 blocks)
    k_gemm_tdm<FIN, 2 * DIN, 1><<<8192, 256, 0, stream>>>(x, W_in, xc, sres);
    // 2) causal depthwise conv + SiLU
    k_conv_silu<<<(R_TOT * DIN) / 256, 256, 0, stream>>>(xc, W_conv, b_conv, u);
    // 3) x_dbl = u @ W_xproj   (4 N-tiles x 128 M-groups = 512 blocks)
    k_gemm_tdm<DIN, 64, 0><<<512, 256, 0, stream>>>(u, W_xprj, xdbl, nullptr);
    // 4) delta + softplus + collapsed delta_p
    k_delta<<<R_TOT / 16, 256, 0, stream>>>(xdbl, W_dt, b_dt, adj, dp);
    // 5) SSM scan (one block per sequence)
    k_scan<<<NSEQ, 256, 0, stream>>>(dp, u, xdbl, sres, A_log, Dv, yact);
    // 6) out = yact @ W_out
    k_gemm_tdm<DIN, FOUT, 0><<<512, 256, 0, stream>>>(yact, W_out, out, nullptr);
}